// QueryGrouper_1717986918814
// MI455X (gfx1250) — compile-verified
//
#include <hip/hip_runtime.h>
#include <stdint.h>

// Problem constants (from reference setup_inputs)
#define B_  8
#define N_  8192
#define M_  2048
#define C_  64
#define K_  64
#define R2_ 0.01f   // RADIUS^2

typedef __attribute__((ext_vector_type(2))) float v2f;
typedef __attribute__((ext_vector_type(8))) float v8f;

// --------------------------------------------------------------------------
// Kernel 1: ball query via V_WMMA_F32_16X16X4_F32.
// One wave (32 threads) owns a tile of 16 query points and streams the 8192
// source points in 16-wide tiles, in ascending index order (the order the
// reference's top_k-of-indices selection requires).
//
// d2 = |q|^2 + |p|^2 - 2 q.p as a single WMMA:
//   A[16x4]  = [qx qy qz 1]           (per query row)
//   B[4x16]  = [-2px -2py -2pz |p|^2] (per point column)
//   C[16x16] = |q|^2 broadcast along each row
//
// Per-row hit masks come straight from ballots: D[v] holds row v (lanes 0-15)
// and row v+8 (lanes 16-31), so ballot(D[v] < r^2) = {row v+8 mask, row v mask}.
// No LDS round-trip, no barriers in the hot loop.
// --------------------------------------------------------------------------
__global__ __launch_bounds__(32) void ball_query_wmma(
    const float* __restrict__ new_xyz,   // (B, 3, M)
    const float* __restrict__ xyz,       // (B, 3, N)
    int*         __restrict__ idx)       // (B, M, K)
{
    const int b    = blockIdx.x / (M_ / 16);
    const int m0   = (blockIdx.x % (M_ / 16)) * 16;
    const int lane = threadIdx.x;        // 0..31, wave32
    const int lm   = lane & 15;
    const bool hi  = lane >= 16;

    __shared__ float lds_qn[16];

    // ---- A matrix + query norms (loop invariant) ----
    const float* nx = new_xyz + (size_t)b * 3 * M_;
    const float qx = nx[0 * M_ + m0 + lm];
    const float qy = nx[1 * M_ + m0 + lm];
    const float qz = nx[2 * M_ + m0 + lm];
    if (!hi) lds_qn[lm] = qx * qx + qy * qy + qz * qz;
    __syncthreads();

    // ISA A layout (16x4 f32): lanes 0-15 hold K=0(V0),K=1(V1); lanes 16-31 K=2,K=3
    v2f A;
    A.x = hi ? qz : qx;
    A.y = hi ? 1.0f : qy;

    // ISA C/D layout: VGPR v -> row v (lanes 0-15) / row v+8 (lanes 16-31)
    v8f Cm;
#pragma unroll
    for (int v = 0; v < 8; ++v) Cm[v] = lds_qn[v + (hi ? 8 : 0)];

    const float* xb = xyz + (size_t)b * 3 * N_;

    int  cnt   = hi ? (1 << 30) : 0;     // hi lanes always look "done" for ballot
    int  first = 0;
    int* idxp  = idx + ((size_t)b * M_ + (m0 + lm)) * K_;   // valid for lanes < 16

    const int rsel = lane;               // lane r < 16 owns query row r

    for (int nt = 0; nt < N_ / 16; ++nt) {
        const int n0 = nt * 16;

        // ---- B matrix tile: 3 coalesced global loads per lane ----
        const float px = xb[0 * N_ + n0 + lm];
        const float py = xb[1 * N_ + n0 + lm];
        const float pz = xb[2 * N_ + n0 + lm];
        const float pn = px * px + py * py + pz * pz;
        // ISA B layout (4x16): V0 = K0 (lanes 0-15) / K2 (lanes 16-31); V1 = K1/K3
        v2f Bm;
        Bm.x = hi ? -2.0f * pz : -2.0f * px;
        Bm.y = hi ? pn         : -2.0f * py;

        // D = A*B + C  ->  full 16x16 tile of squared distances
        v8f D = __builtin_amdgcn_wmma_f32_16x16x4_f32(
            /*neg_a=*/false, A, /*neg_b=*/false, Bm,
            /*c_mod=*/(short)0, Cm, /*reuse_a=*/false, /*reuse_b=*/false);

        // ---- per-row masks via ballot: bits[15:0]=row v, bits[31:16]=row v+8 ----
        uint32_t b0 = __builtin_amdgcn_ballot_w32(D[0] < R2_);
        uint32_t b1 = __builtin_amdgcn_ballot_w32(D[1] < R2_);
        uint32_t b2 = __builtin_amdgcn_ballot_w32(D[2] < R2_);
        uint32_t b3 = __builtin_amdgcn_ballot_w32(D[3] < R2_);
        uint32_t b4 = __builtin_amdgcn_ballot_w32(D[4] < R2_);
        uint32_t b5 = __builtin_amdgcn_ballot_w32(D[5] < R2_);
        uint32_t b6 = __builtin_amdgcn_ballot_w32(D[6] < R2_);
        uint32_t b7 = __builtin_amdgcn_ballot_w32(D[7] < R2_);

        // lane r picks ballot[r & 7]; rows 0-7 take the low half, 8-15 the high half
        uint32_t s01   = (rsel & 1) ? b1 : b0;
        uint32_t s23   = (rsel & 1) ? b3 : b2;
        uint32_t s45   = (rsel & 1) ? b5 : b4;
        uint32_t s67   = (rsel & 1) ? b7 : b6;
        uint32_t s0123 = (rsel & 2) ? s23 : s01;
        uint32_t s4567 = (rsel & 2) ? s67 : s45;
        uint32_t sel   = (rsel & 4) ? s4567 : s0123;
        uint32_t m16   = (rsel & 8) ? (sel >> 16) : (sel & 0xFFFFu);
        if (hi) m16 = 0;                 // hi lanes are column helpers, not rows

        // ---- extract hit indices in ascending order (rarely non-empty) ----
        while (m16) {
            const int t = __builtin_ctz(m16);
            m16 &= m16 - 1;
            const int n = n0 + t;
            if (cnt == 0)  first = n;
            if (cnt < K_)  idxp[cnt] = n;
            ++cnt;
        }

        // uniform early exit once every row has K_ hits
        const uint32_t done = __builtin_amdgcn_ballot_w32(cnt >= K_);
        if (done == 0xFFFFFFFFu) break;
    }

    // pad unfilled slots with first hit (or 0 if none) — matches reference
    if (!hi) {
        const int c = cnt < K_ ? cnt : K_;
        for (int k = c; k < K_; ++k) idxp[k] = first;
    }
}

// --------------------------------------------------------------------------
// Kernel 2: gather + concat. One 64-thread block per (b, m); thread k owns
// neighbor slot k. Writes are 64-wide contiguous (fully coalesced per wave)
// and non-temporal (output is 281 MB write-once, > 192 MB L2); gathered reads
// stay L2-resident (feature is only 16 MB).
// Output layout: (B, 3 + C, M, K), grouped_xyz first, then grouped features.
// --------------------------------------------------------------------------
__global__ __launch_bounds__(64) void group_gather(
    const float* __restrict__ new_xyz,   // (B, 3, M)
    const float* __restrict__ xyz,       // (B, 3, N)
    const float* __restrict__ feature,   // (B, C, N)
    const int*   __restrict__ idx,       // (B, M, K)
    float*       __restrict__ out)       // (B, 3+C, M, K)
{
    const int bm = blockIdx.x;
    const int b  = bm / M_;
    const int m  = bm % M_;
    const int k  = threadIdx.x;

    const int id = idx[((size_t)b * M_ + m) * K_ + k];

    const float* xb = xyz     + (size_t)b * 3 * N_;
    const float* nb = new_xyz + (size_t)b * 3 * M_;
    const float* fb = feature + (size_t)b * C_ * N_;
    float* ob = out + (size_t)b * (3 + C_) * M_ * K_ + (size_t)m * K_ + k;

#pragma unroll
    for (int ch = 0; ch < 3; ++ch)
        __builtin_nontemporal_store(xb[ch * N_ + id] - nb[ch * M_ + m],
                                    &ob[(size_t)ch * (M_ * K_)]);

#pragma unroll 4
    for (int ch = 0; ch < C_; ++ch)
        __builtin_nontemporal_store(fb[ch * N_ + id],
                                    &ob[(size_t)(3 + ch) * (M_ * K_)]);
}

// --------------------------------------------------------------------------
extern "C" void kernel_launch(void* const* d_in, const int* in_sizes, int n_in,
                              void* d_out, int out_size, void* d_ws, size_t ws_size,
                              hipStream_t stream) {
    (void)in_sizes; (void)n_in; (void)out_size; (void)ws_size;

    const float* new_xyz = (const float*)d_in[0];   // (8, 3, 2048)
    const float* xyz     = (const float*)d_in[1];   // (8, 3, 8192)
    const float* feature = (const float*)d_in[2];   // (8, 64, 8192)
    float* out = (float*)d_out;                     // (8, 67, 2048, 64)
    int*   idx = (int*)d_ws;                        // (8, 2048, 64) = 4 MB scratch

    ball_query_wmma<<<B_ * (M_ / 16), 32, 0, stream>>>(new_xyz, xyz, idx);
    group_gather<<<B_ * M_, 64, 0, stream>>>(new_xyz, xyz, feature, idx, out);
}